// PatchExtractor_53197464928812
// MI455X (gfx1250) — compile-verified
//
#include <hip/hip_runtime.h>

typedef __attribute__((ext_vector_type(2))) float v2f;
typedef __attribute__((ext_vector_type(8))) float v8f;

// Problem constants (from reference)
#define PC    32
#define PH    96           // H = W = D = 96
#define PP    15
#define KDIM  108000       // C * P^3
#define MDIM  128          // B * N
#define FDIM  1024

// GEMM tiling
#define BK      16
#define BKP     20         // A LDS pitch: 20m mod 64 distinct, spacing 4 -> conflict-free b64 frags
#define BN      256
#define P2      544        // B LDS pitch per k-pair: 544 % 64 == 32 banks -> hi/lo half-waves disjoint
#define KSPLIT  45
#define KCHUNK  (KDIM / KSPLIT)   // 2400
#define KTILES  (KCHUNK / BK)     // 150

// ---------------------------------------------------------------------------
__global__ void init_out_kernel(float* __restrict__ out, const float* __restrict__ bias) {
  int i = blockIdx.x * blockDim.x + threadIdx.x;
  if (i < MDIM * FDIM) out[i] = bias[i & (FDIM - 1)];
}

// ---------------------------------------------------------------------------
// Materialize A = flattened patches, (MDIM x KDIM) row-major in d_ws.
__global__ void extract_patches_kernel(const float* __restrict__ x,
                                       const int* __restrict__ cen,
                                       float* __restrict__ A) {
  long long idx = (long long)blockIdx.x * blockDim.x + threadIdx.x;
  const long long total = (long long)MDIM * KDIM;
  if (idx >= total) return;
  int p = (int)(idx / KDIM);
  int k = (int)(idx % KDIM);
  int b = p >> 6;
  int c   = k / (PP * PP * PP);
  int rem = k % (PP * PP * PP);
  int rx  = rem / (PP * PP);
  int ry  = (rem / PP) % PP;
  int rz  = rem % PP;

  int cx = cen[p * 3 + 0];
  int cy = cen[p * 3 + 1];
  int cz = cen[p * 3 + 2];

  int sx = max(cx - 7, 0), ex = min(cx + 8, PH);
  int sy = max(cy - 7, 0), ey = min(cy + 8, PH);
  int sz = max(cz - 7, 0), ez = min(cz + 8, PH);

  int ix = sx + rx, iy = sy + ry, iz = sz + rz;
  bool ok = (ix < ex) && (iy < ey) && (iz < ez);
  ix = min(ix, PH - 1); iy = min(iy, PH - 1); iz = min(iz, PH - 1);

  float v = 0.0f;
  if (ok) {
    int vi = (((b * PC + c) * PH + ix) * PH + iy) * PH + iz;
    v = x[vi];
  }
  A[idx] = v;
}

// ---------------------------------------------------------------------------
// Async DMA global -> LDS (GVS mode), ASYNCcnt-tracked.
__device__ __forceinline__ void async_b128(const void* sbase, unsigned goff_bytes,
                                           unsigned lds_addr) {
  asm volatile("global_load_async_to_lds_b128 %0, %1, %2"
               :: "v"(lds_addr), "v"(goff_bytes), "s"(sbase)
               : "memory");
}
__device__ __forceinline__ void async_b32(const void* sbase, unsigned goff_bytes,
                                          unsigned lds_addr) {
  asm volatile("global_load_async_to_lds_b32 %0, %1, %2"
               :: "v"(lds_addr), "v"(goff_bytes), "s"(sbase)
               : "memory");
}

// ---------------------------------------------------------------------------
// C(128x1024) += A(128xK) * W(KxF).  Split-K, double-buffered LDS fed by
// GLOBAL_LOAD_ASYNC_TO_LDS, f32 WMMA 16x16x4.
// Block: 512 threads = 16 waves; block tile 128 x 256 over one K chunk.
// Wave (mg = wave&3, ng = wave>>2) owns rows [mg*32,+32) x cols [ng*64,+64).
//
// B LDS layout is k-pair interleaved: (k, n) -> word (k>>1)*P2 + 2n + (k&1),
// so every WMMA B fragment (k, k+1 at fixed n) is one ds_load_b64 into an
// even-aligned VGPR pair (no repacking), and adjacent n-tiles (128 B apart)
// fuse into ds_load_2addr_b64.
__global__ __launch_bounds__(512) void gemm_kernel(const float* __restrict__ A,
                                                   const float* __restrict__ Wg,
                                                   float* __restrict__ out) {
  __shared__ float As[2][MDIM * BKP];        // [m][k]           10240 B per buffer
  __shared__ float Bs[2][(BK / 2) * P2];     // pair-interleaved 17408 B per buffer

  const int tid    = threadIdx.x;
  const int nbase  = blockIdx.x * BN;
  const int kg0    = blockIdx.y * KCHUNK;
  const int wave   = tid >> 5;
  const int lane   = tid & 31;
  const int mg     = wave & 3;
  const int ng     = wave >> 2;
  const int laneLo = lane & 15;
  const int hi     = lane >> 4;   // lanes 16-31: K+2..3 (A/B frags), M+8 (C rows)

  // ---- A staging: one async b128 per thread per tile (4 float4 per row).
  const int am = tid >> 2;
  const int aq = tid & 3;
  const unsigned a_lds[2] = {
    (unsigned)(uintptr_t)&As[0][am * BKP + aq * 4],
    (unsigned)(uintptr_t)&As[1][am * BKP + aq * 4] };
  const unsigned a_gbase = (unsigned)((am * KDIM + aq * 4) * 4);

  // ---- B staging: 8 async b32 per thread per tile.
  // Thread covers n = tid&255 at rows r0, r0+2, ..., r0+14 (r0 = tid>>8).
  // LDS word for step j: j*P2 + 2n + r0  -> constant stride P2*4 bytes.
  // Global byte for step j: ((kg + r0 + 2j)*FDIM + nbase + n)*4 -> stride 8192.
  const int bn = tid & 255;
  const int r0 = tid >> 8;
  const unsigned b_lds[2] = {
    (unsigned)(uintptr_t)&Bs[0][2 * bn + r0],
    (unsigned)(uintptr_t)&Bs[1][2 * bn + r0] };
  const unsigned b_gbase = (unsigned)((r0 * FDIM + nbase + bn) * 4);

  v8f acc[2][4] = {};

  // Prologue: kick DMA for tile 0 into buffer 0.
  {
    const unsigned kb = (unsigned)(kg0 * 4);
    async_b128(A, a_gbase + kb, a_lds[0]);
#pragma unroll
    for (int j = 0; j < 8; ++j)
      async_b32(Wg, b_gbase + kb * FDIM + j * 8192u, b_lds[0] + j * (P2 * 4u));
  }

  for (int t = 0; t < KTILES; ++t) {
    const int buf = t & 1;

    if (t + 1 < KTILES) {
      // Kick DMA for tile t+1 into the other buffer (its readers finished at
      // the end of iteration t-1, behind a barrier).
      const unsigned kb = (unsigned)((kg0 + (t + 1) * BK) * 4);
      async_b128(A, a_gbase + kb, a_lds[buf ^ 1]);
#pragma unroll
      for (int j = 0; j < 8; ++j)
        async_b32(Wg, b_gbase + kb * FDIM + j * 8192u, b_lds[buf ^ 1] + j * (P2 * 4u));
      // Async loads complete in order: <=9 pending => tile t's 9 loads landed.
      asm volatile("s_wait_asynccnt 0x9" ::: "memory");
    } else {
      asm volatile("s_wait_asynccnt 0x0" ::: "memory");
    }
    __syncthreads();   // all waves' tile-t DMA visible to everyone

    const float* as = As[buf];
    const float* bs = Bs[buf];

#pragma unroll
    for (int ks = 0; ks < 4; ++ks) {
      const int kb  = ks * 4 + hi * 2;    // A k offset within tile
      const int kbp = ks * 2 + hi;        // B k-pair index
      const int n0  = ng * 64 + laneLo;
      v2f a0 = *(const v2f*)&as[(mg * 32 +      laneLo) * BKP + kb];
      v2f a1 = *(const v2f*)&as[(mg * 32 + 16 + laneLo) * BKP + kb];
      v2f b0 = *(const v2f*)&bs[kbp * P2 + 2 * (n0     )];
      v2f b1 = *(const v2f*)&bs[kbp * P2 + 2 * (n0 + 16)];
      v2f b2 = *(const v2f*)&bs[kbp * P2 + 2 * (n0 + 32)];
      v2f b3 = *(const v2f*)&bs[kbp * P2 + 2 * (n0 + 48)];

      acc[0][0] = __builtin_amdgcn_wmma_f32_16x16x4_f32(false, a0, false, b0, (short)0, acc[0][0], false, false);
      acc[0][1] = __builtin_amdgcn_wmma_f32_16x16x4_f32(false, a0, false, b1, (short)0, acc[0][1], false, false);
      acc[0][2] = __builtin_amdgcn_wmma_f32_16x16x4_f32(false, a0, false, b2, (short)0, acc[0][2], false, false);
      acc[0][3] = __builtin_amdgcn_wmma_f32_16x16x4_f32(false, a0, false, b3, (short)0, acc[0][3], false, false);
      acc[1][0] = __builtin_amdgcn_wmma_f32_16x16x4_f32(false, a1, false, b0, (short)0, acc[1][0], false, false);
      acc[1][1] = __builtin_amdgcn_wmma_f32_16x16x4_f32(false, a1, false, b1, (short)0, acc[1][1], false, false);
      acc[1][2] = __builtin_amdgcn_wmma_f32_16x16x4_f32(false, a1, false, b2, (short)0, acc[1][2], false, false);
      acc[1][3] = __builtin_amdgcn_wmma_f32_16x16x4_f32(false, a1, false, b3, (short)0, acc[1][3], false, false);
    }

    __syncthreads();   // everyone done reading buf before it is re-filled
  }

  // Split-K reduction: native f32 atomics into bias-initialized out.
  // C/D layout: VGPR v holds M = v (lanes 0-15) / v+8 (lanes 16-31), N = laneLo.
#pragma unroll
  for (int i = 0; i < 2; ++i) {
#pragma unroll
    for (int j = 0; j < 4; ++j) {
      int mbase = mg * 32 + i * 16 + hi * 8;
      int ncol  = nbase + ng * 64 + j * 16 + laneLo;
#pragma unroll
      for (int v = 0; v < 8; ++v) {
        __hip_atomic_fetch_add(&out[(mbase + v) * FDIM + ncol], acc[i][j][v],
                               __ATOMIC_RELAXED, __HIP_MEMORY_SCOPE_AGENT);
      }
    }
  }
}

// ---------------------------------------------------------------------------
extern "C" void kernel_launch(void* const* d_in, const int* in_sizes, int n_in,
                              void* d_out, int out_size, void* d_ws, size_t ws_size,
                              hipStream_t stream) {
  const float* x    = (const float*)d_in[0];   // (2,32,96,96,96) f32
  const int*   cen  = (const int*)d_in[1];     // (2,64,3) i32
  const float* Wm   = (const float*)d_in[2];   // (108000,1024) f32
  const float* bias = (const float*)d_in[3];   // (1024,) f32
  float* out  = (float*)d_out;                 // (2,64,1024) f32
  float* Apat = (float*)d_ws;                  // 128*108000 f32 = 55.3 MB scratch

  init_out_kernel<<<(MDIM * FDIM + 255) / 256, 256, 0, stream>>>(out, bias);

  long long total = (long long)MDIM * KDIM;
  int eblocks = (int)((total + 255) / 256);
  extract_patches_kernel<<<eblocks, 256, 0, stream>>>(x, cen, Apat);

  dim3 grid(FDIM / BN, KSPLIT);
  gemm_kernel<<<grid, 512, 0, stream>>>(Apat, Wm, out);
}